// _RFFTDecomp_2748779070249
// MI455X (gfx1250) — compile-verified
//
#include <hip/hip_runtime.h>
#include <math.h>

// CDNA5 wave32 WMMA fragment types: 16x16x4 f32 -> A,B = 2 VGPRs, C/D = 8 VGPRs
typedef float v2f __attribute__((ext_vector_type(2)));
typedef float v8f __attribute__((ext_vector_type(8)));

#define NFFT   1024
#define BASEN  16            // base DFT size (WMMA tile N=K=16)
#define NCHUNK 64            // 1024 / 16 decimated chunks, log2 = 6
#define WAVES  4             // waves (rows) per workgroup
#define PI_F   3.14159265358979323846f

__device__ __forceinline__ int brev6(int c) { return (int)(__brev((unsigned)c) >> 26); }

__global__ __launch_bounds__(WAVES * 32) void rfft1024_kernel(
    const float* __restrict__ x, float* __restrict__ out, int nrows) {
  // Per-wave: sx = staged input row (aliased region), ur/ui = real/imag work arrays.
  __shared__ __align__(16) float buf[WAVES][3 * NFFT];   // 48 KB
  __shared__ float BRt[BASEN * BASEN];                   // 16-pt DFT real:  cos(-2pi kn/16)
  __shared__ float BIt[BASEN * BASEN];                   // 16-pt DFT imag:  sin(-2pi kn/16)
  __shared__ float WRt[512];                             // master twiddles exp(-i pi k/512)
  __shared__ float WIt[512];

  const int tid = threadIdx.x;

  // ---- one-time per-WG tables (amortized over 4 rows) ----
  for (int i = tid; i < BASEN * BASEN; i += WAVES * 32) {
    const int k = i >> 4, n = i & 15;
    const float ang = -2.0f * PI_F * (float)((k * n) & 15) * (1.0f / 16.0f);
    BRt[i] = cosf(ang);
    BIt[i] = sinf(ang);
  }
  for (int i = tid; i < 512; i += WAVES * 32) {
    const float ang = -PI_F * (float)i * (1.0f / 512.0f);
    WRt[i] = cosf(ang);
    WIt[i] = sinf(ang);
  }
  __syncthreads();

  const int wave = tid >> 5;
  const int lane = tid & 31;
  const int half = lane >> 4;      // wave32 half-wave select for WMMA K-lanes
  const int m16  = lane & 15;
  const int row  = blockIdx.x * WAVES + wave;       // grid exactly covers nrows

  float* sx = buf[wave];
  float* ur = sx + NFFT;
  float* ui = ur + NFFT;

  // ---- stage the 4 KB input row into LDS, fully coalesced b128 ----
  const float4* xr4 = (const float4*)(x + (size_t)row * NFFT);
  float4* sx4 = (float4*)sx;
#pragma unroll
  for (int i = 0; i < NFFT / (32 * 4); ++i) sx4[lane + 32 * i] = xr4[lane + 32 * i];
  __syncthreads();

  // ---- base-16 DFT of 64 decimated chunks via fp32 WMMA ----
  // chunk c = { x[64*j + brev6(c)] : j=0..15 }  (replaces the PERM matmul)
  // A[m][k] = sx[64*k + brev6(16*mt+m)], B_R[k][n] = cos(-2pi kn/16), B_I = sin.
#pragma unroll
  for (int mt = 0; mt < NCHUNK / 16; ++mt) {
    const int cOff = brev6(mt * 16 + m16);
    v8f accR = {0.f, 0.f, 0.f, 0.f, 0.f, 0.f, 0.f, 0.f};
    v8f accI = {0.f, 0.f, 0.f, 0.f, 0.f, 0.f, 0.f, 0.f};
#pragma unroll
    for (int t = 0; t < 4; ++t) {               // K = 16 -> 4 steps of K=4
      const int k0 = 4 * t + 2 * half;          // ISA f32 A layout: lanes<16 K=+0,+1; lanes>=16 K=+2,+3
      v2f a, bR, bI;
      a[0]  = sx[(k0 + 0) * NCHUNK + cOff];
      a[1]  = sx[(k0 + 1) * NCHUNK + cOff];
      bR[0] = BRt[(k0 + 0) * 16 + m16];
      bR[1] = BRt[(k0 + 1) * 16 + m16];
      bI[0] = BIt[(k0 + 0) * 16 + m16];
      bI[1] = BIt[(k0 + 1) * 16 + m16];
      accR = __builtin_amdgcn_wmma_f32_16x16x4_f32(false, a, false, bR, (short)0, accR, false, false);
      accI = __builtin_amdgcn_wmma_f32_16x16x4_f32(false, a, false, bI, (short)0, accI, false, false);
    }
    // C/D layout: VGPR i holds M = i (lanes 0-15) / i+8 (lanes 16-31), N = lane&15
#pragma unroll
    for (int i = 0; i < 8; ++i) {
      const int c = mt * 16 + i + 8 * half;
      ur[c * 16 + m16] = accR[i];
      ui[c * 16 + m16] = accI[i];
    }
  }
  __syncthreads();

  // ---- 6 radix-2 combine stages in LDS (VALU fp32, table twiddles) ----
  // stage s: L = 16<<s, merge chunk pairs (2g,2g+1); w_k = exp(-i pi k / L) = master[k << (5-s)]
  for (int s = 0; s < 6; ++s) {
    const int L = 16 << s;
    const int tsh = 5 - s;
    for (int b = lane; b < 512; b += 32) {
      const int g  = b >> (4 + s);
      const int k  = b & (L - 1);
      const int i0 = (g << (5 + s)) + k;
      const int i1 = i0 + L;
      const float er = ur[i0], ei = ui[i0];
      const float orr = ur[i1], oii = ui[i1];
      const float wr = WRt[k << tsh], wi = WIt[k << tsh];
      const float tr = wr * orr - wi * oii;
      const float ti = wr * oii + wi * orr;
      ur[i0] = er + tr;  ui[i0] = ei + ti;
      ur[i1] = er - tr;  ui[i1] = ei - ti;
    }
    __syncthreads();
  }

  // ---- emit bins 0..512: d_out = [real (nrows x 513) | imag (nrows x 513)] ----
  float* outR = out + (size_t)row * 513;
  float* outI = out + (size_t)nrows * 513 + (size_t)row * 513;
  for (int k = lane; k < 513; k += 32) {
    outR[k] = ur[k];
    outI[k] = ui[k];
  }
}

extern "C" void kernel_launch(void* const* d_in, const int* in_sizes, int n_in,
                              void* d_out, int out_size, void* d_ws, size_t ws_size,
                              hipStream_t stream) {
  const float* x = (const float*)d_in[0];
  float* out = (float*)d_out;
  const int nrows = in_sizes[0] / NFFT;          // 16*512 = 8192
  const int blocks = nrows / WAVES;              // 2048 blocks * 4 waves = 8192 rows
  rfft1024_kernel<<<blocks, WAVES * 32, 0, stream>>>(x, out, nrows);
}